// GraphRecommender_7060926234755
// MI455X (gfx1250) — compile-verified
//
#include <hip/hip_runtime.h>

typedef __attribute__((ext_vector_type(2))) float v2f;
typedef __attribute__((ext_vector_type(8))) float v8f;

namespace {
constexpr int  kNU1 = 100001;          // users + 1
constexpr int  kNI1 = 50001;           // items + 1
constexpr int  kN   = kNU1 + kNI1;     // 150002 nodes
constexpr int  kH   = 128;             // hidden dim
constexpr long kE   = 1500000;         // original edges (undirected -> 2*kE directed)
}

__device__ __forceinline__ void atomAddF(float* p, float v) {
  __hip_atomic_fetch_add(p, v, __ATOMIC_RELAXED, __HIP_MEMORY_SCOPE_AGENT);
}

// ---- x = concat(user_emb, item_emb), vectorized float4 ----
__global__ void gr_concat_kern(const float4* __restrict__ user,
                               const float4* __restrict__ item,
                               float4* __restrict__ x, long n4) {
  long i = (long)blockIdx.x * blockDim.x + threadIdx.x;
  const long nu4 = (long)kNU1 * kH / 4;
  if (i < n4) x[i] = (i < nu4) ? user[i] : item[i - nu4];
}

__global__ void gr_zero_kern(float* __restrict__ p, long n) {
  long i = (long)blockIdx.x * blockDim.x + threadIdx.x;
  if (i < n) p[i] = 0.0f;
}

__global__ void gr_zero4_kern(float4* __restrict__ p, long n4) {
  long i = (long)blockIdx.x * blockDim.x + threadIdx.x;
  if (i < n4) p[i] = make_float4(0.f, 0.f, 0.f, 0.f);
}

// ---- undirected degree count (float accumulation; exact for small counts) ----
__global__ void gr_degree_kern(const int* __restrict__ ei, float* __restrict__ deg) {
  long e = (long)blockIdx.x * blockDim.x + threadIdx.x;
  if (e < kE) {
    int s = ei[e];
    int d = ei[kE + e];
    atomAddF(&deg[s], 1.0f);
    atomAddF(&deg[d], 1.0f);
  }
}

__global__ void gr_invdeg_kern(float* __restrict__ deg, long n) {
  long i = (long)blockIdx.x * blockDim.x + threadIdx.x;
  if (i < n) {
    float v = deg[i];
    deg[i] = (v > 0.0f) ? (1.0f / v) : 0.0f;
  }
}

// ---- scatter-add: agg[dst] += x[src] over both edge directions ----
// one wave (32 lanes) per directed edge; lane handles 4 contiguous floats (b128 load).
// edge id is wave-uniform -> readfirstlane so index fetches ride the scalar path.
__global__ void gr_scatter_kern(const int* __restrict__ ei,
                                const float* __restrict__ x,
                                float* __restrict__ agg) {
  long de = (long)blockIdx.x * (blockDim.x >> 5) +
            (long)__builtin_amdgcn_readfirstlane(threadIdx.x >> 5);
  int lane = (int)(threadIdx.x & 31);
  if (de >= 2L * kE) return;
  int s, d;
  if (de < kE) { s = ei[de];       d = ei[kE + de]; }
  else         { long e = de - kE; d = ei[e]; s = ei[kE + e]; }
  const float4 v = *(const float4*)(x + (long)s * kH + lane * 4);
  float* ap = agg + (long)d * kH + lane * 4;
  atomAddF(ap + 0, v.x);
  atomAddF(ap + 1, v.y);
  atomAddF(ap + 2, v.z);
  atomAddF(ap + 3, v.w);
}

// ---- fused SAGE layer: xout = relu( (agg*invdeg) @ Wl^T + b + xin @ Wr^T ) ----
// grid.x = ceil(N/16) row blocks; 8 waves per block, wave w owns output cols [16w,16w+16)
// V_WMMA_F32_16X16X4_F32: A 16x4 (v2f/lane), B 4x16 (v2f/lane), C/D 16x16 (v8f/lane)
// Row-validity is folded into multiplicative masks so the K-loop has NO divergent
// branches: 4x global_load_b64 + 2x v_pk_mul_f32 + 2x v_wmma per K-step.
__global__ void gr_sage_gemm_kern(const float* __restrict__ xin,
                                  const float* __restrict__ agg,
                                  const float* __restrict__ invdeg,
                                  const float* __restrict__ Wl,
                                  const float* __restrict__ Wr,
                                  const float* __restrict__ bias,
                                  float* __restrict__ xout) {
  const int lane = threadIdx.x & 31;
  const int w    = threadIdx.x >> 5;       // wave id 0..7 -> column tile
  const int lr   = lane & 15;
  const int half = lane >> 4;              // selects K-pair (A/B)
  const int row0 = blockIdx.x * 16;

  const int   arow   = row0 + lr;                      // A-tile row for this lane
  const bool  rvalid = (arow < kN);
  const int   arowc  = rvalid ? arow : (kN - 1);       // clamped: loads always legal
  const float vmask  = rvalid ? 1.0f : 0.0f;
  const float sc     = invdeg[arowc] * vmask;          // mean-norm + validity mask

  const float* aggRow = agg + (long)arowc * kH;
  const float* xRow   = xin + (long)arowc * kH;
  const int    ncol   = w * 16 + lr;                   // global output column

  v8f c;
  {
    const float b = bias[ncol];
#pragma unroll
    for (int j = 0; j < 8; ++j) c[j] = b;
  }

  const float* WlRow = Wl + (long)ncol * kH;           // W[n][k] row-major
  const float* WrRow = Wr + (long)ncol * kH;

#pragma unroll 8
  for (int k0 = 0; k0 < kH; k0 += 4) {
    const int ka = k0 + half * 2;
    v2f a1 = *(const v2f*)(aggRow + ka) * sc;
    v2f a2 = *(const v2f*)(xRow + ka) * vmask;
    const v2f b1 = *(const v2f*)(WlRow + ka);
    const v2f b2 = *(const v2f*)(WrRow + ka);
    c = __builtin_amdgcn_wmma_f32_16x16x4_f32(false, a1, false, b1,
                                              (short)0, c, false, false);
    c = __builtin_amdgcn_wmma_f32_16x16x4_f32(false, a2, false, b2,
                                              (short)0, c, false, false);
  }

  // C/D layout: VGPR j, lanes 0-15: M=j ; lanes 16-31: M=j+8 ; N = lane%16
#pragma unroll
  for (int j = 0; j < 8; ++j) {
    const int row = row0 + j + half * 8;
    if (row < kN) {
      float v = c[j];
      xout[(long)row * kH + ncol] = (v > 0.0f) ? v : 0.0f;
    }
  }
}

extern "C" void kernel_launch(void* const* d_in, const int* in_sizes, int n_in,
                              void* d_out, int out_size, void* d_ws, size_t ws_size,
                              hipStream_t stream) {
  const float* user = (const float*)d_in[0];
  const float* item = (const float*)d_in[1];
  const float* Wl   = (const float*)d_in[2];  // [2,128,128]
  const float* bl   = (const float*)d_in[3];  // [2,128]
  const float* Wr   = (const float*)d_in[4];  // [2,128,128]
  const int*   ei   = (const int*)d_in[5];    // [2,E] int32
  float* out = (float*)d_out;                 // [N,128]

  // workspace layout: x (N*128) | agg (N*128) | deg (N)   ~154 MB total
  float* x   = (float*)d_ws;
  float* agg = x + (size_t)kN * kH;
  float* deg = agg + (size_t)kN * kH;

  const long nfeat = (long)kN * kH;
  const long n4    = nfeat / 4;
  const int  T     = 256;

  // node features
  gr_concat_kern<<<(unsigned)((n4 + T - 1) / T), T, 0, stream>>>(
      (const float4*)user, (const float4*)item, (float4*)x, n4);

  // inverse degrees (stored in-place in deg)
  gr_zero_kern<<<(unsigned)((kN + T - 1) / T), T, 0, stream>>>(deg, kN);
  gr_degree_kern<<<(unsigned)((kE + T - 1) / T), T, 0, stream>>>(ei, deg);
  gr_invdeg_kern<<<(unsigned)((kN + T - 1) / T), T, 0, stream>>>(deg, kN);

  const unsigned gemmGrid = (unsigned)((kN + 15) / 16);
  const long     scatWaves   = 2L * kE;                 // one wave per directed edge
  const unsigned scatGrid    = (unsigned)((scatWaves + 7) / 8);  // 8 waves / 256-thr block
  const unsigned zeroGrid    = (unsigned)((n4 + T - 1) / T);

  // ---- layer 0: reads x, writes out ----
  gr_zero4_kern<<<zeroGrid, T, 0, stream>>>((float4*)agg, n4);
  gr_scatter_kern<<<scatGrid, T, 0, stream>>>(ei, x, agg);
  gr_sage_gemm_kern<<<gemmGrid, T, 0, stream>>>(x, agg, deg,
                                                Wl, Wr, bl, out);

  // ---- layer 1: reads out, writes x ----
  gr_zero4_kern<<<zeroGrid, T, 0, stream>>>((float4*)agg, n4);
  gr_scatter_kern<<<scatGrid, T, 0, stream>>>(ei, out, agg);
  gr_sage_gemm_kern<<<gemmGrid, T, 0, stream>>>(out, agg, deg,
                                                Wl + 128 * 128, Wr + 128 * 128,
                                                bl + 128, x);

  // final result -> d_out (graph-capture-safe async D2D copy)
  hipMemcpyAsync(out, x, (size_t)nfeat * sizeof(float),
                 hipMemcpyDeviceToDevice, stream);
}